// Homography_Loss_69647189672373
// MI455X (gfx1250) — compile-verified
//
#include <hip/hip_runtime.h>
#include <math.h>

typedef float v2f __attribute__((ext_vector_type(2)));
typedef float v8f __attribute__((ext_vector_type(8)));

__device__ const float CLS_MEAN[3][3] = {
    {1.76255119f, 0.66068622f, 0.84422524f},
    {1.52563191462f, 1.62856739989f, 3.88311640418f},
    {1.73698127f, 0.59706367f, 1.76282397f}};

// ---------------------------------------------------------------------------
// Kernel 1: per-object geometry -> per-variant dst (x,z) points (5 per object)
// ---------------------------------------------------------------------------
__global__ void build_dst_kernel(
    const float* __restrict__ center, const int* __restrict__ cls_ids,
    const float* __restrict__ ry, const float* __restrict__ size3d,
    const float* __restrict__ off_t, const float* __restrict__ depth_t,
    const float* __restrict__ off_o, const float* __restrict__ depth_o,
    const float* __restrict__ calibs, const float* __restrict__ trans_inv,
    float* __restrict__ dstws, int K, int M) {
  int k = blockIdx.x * blockDim.x + threadIdx.x;
  if (k >= K) return;
  int b = k / M;

  float cx_ = center[2 * k], cy_ = center[2 * k + 1];
  int cls = cls_ids[k];
  float d1 = size3d[3 * k + 1] + CLS_MEAN[cls][1];
  float d2 = size3d[3 * k + 2] + CLS_MEAN[cls][2];
  // d0 only affects Qpred.y, which the loss never reads.

  float sn, cs;
  sincosf(ry[k], &sn, &cs);

  const float* cal = calibs + (size_t)b * 12;
  float fx = cal[0], cx = cal[2], b03 = cal[3];
  float tx = b03 / (-fx);  // y-channel of pts unused downstream
  const float* ti = trans_inv + (size_t)b * 6;

  // corners5: [0, R*xyz_n]; xyz row0 = d2*BASE2, row1 = 0, row2 = d1*BASE1
  const float B2[4] = {0.5f, 0.5f, -0.5f, -0.5f};
  const float B1[4] = {0.5f, -0.5f, -0.5f, 0.5f};
  float c5x[5], c5z[5];
  c5x[0] = 0.0f; c5z[0] = 0.0f;
#pragma unroll
  for (int n = 0; n < 4; n++) {
    float x0 = d2 * B2[n];
    float x2 = d1 * B1[n];
    c5x[n + 1] = cs * x0 + sn * x2;
    c5z[n + 1] = -sn * x0 + cs * x2;
  }

  float ox[3], oy[3], dp[3];
  ox[0] = off_t[2 * k]; oy[0] = off_t[2 * k + 1]; dp[0] = depth_o[2 * k];
  ox[1] = off_o[2 * k]; oy[1] = off_o[2 * k + 1]; dp[1] = depth_t[k];
  ox[2] = off_o[2 * k]; oy[2] = off_o[2 * k + 1]; dp[2] = depth_o[2 * k];

#pragma unroll
  for (int v = 0; v < 3; v++) {
    float ux = (cx_ + ox[v]) * 4.0f;
    float uy = (cy_ + oy[v]) * 4.0f;
    float u0 = ti[0] * ux + ti[1] * uy + ti[2];
    float dep = dp[v];
    float px = (u0 - cx) * dep / fx + tx;
    float pz = dep;
    float* out = dstws + ((size_t)v * K + k) * 10;
#pragma unroll
    for (int p = 0; p < 5; p++) {
      out[2 * p + 0] = c5x[p] + px;   // Qpred.x
      out[2 * p + 1] = c5z[p] + pz;   // Qpred.z
    }
  }
}

// ---------------------------------------------------------------------------
// Kernel 2: one wave per (batch, variant). AtA via v_wmma_f32_16x16x4_f32,
// then lane 0 Jacobi-eigensolves the 9x9 and writes H (min-eigvec).
//
// Branchless DLT fragment build: for a lane with component idx (0..15),
//   even row:  e0 = ce.(x,y,1) - u * (g.(x,y,1))
//   odd  row:  e1 = co.(x,y,1) - w * (g.(x,y,1))
// with ce = 1[idx==0..2], co = 1[idx==3..5], g = 1[idx==6..8]; idx>=9 -> 0.
// Each lane's two K slots are the even/odd rows of point p = cc*2 + half.
// ---------------------------------------------------------------------------
__global__ void __launch_bounds__(32)
ata_wmma_kernel(const float* __restrict__ Pgt, const float* __restrict__ dstws,
                float* __restrict__ Hout, int B, int M) {
  int b = blockIdx.x;
  int v = blockIdx.y;
  int lane = threadIdx.x;  // 0..31, wave32
  const int P = M * 5;     // points per batch (640)

  const float* src = Pgt + (size_t)b * P * 2;
  const float* dst = dstws + ((size_t)v * B + b) * (size_t)P * 2;

  __shared__ float4 pt[16];
  __shared__ float Nsh[16][16];

  v8f acc = {};
  int half = lane >> 4;   // 0/1 -> K offset {0,2}
  int idx = lane & 15;    // Aᵀ row == A col: identical fragment

  // Hoisted per-lane coefficient selects (loop-invariant, branchless).
  float ce_x = (idx == 0) ? 1.0f : 0.0f;
  float ce_y = (idx == 1) ? 1.0f : 0.0f;
  float ce_1 = (idx == 2) ? 1.0f : 0.0f;
  float co_x = (idx == 3) ? 1.0f : 0.0f;
  float co_y = (idx == 4) ? 1.0f : 0.0f;
  float co_1 = (idx == 5) ? 1.0f : 0.0f;
  float g_x  = (idx == 6) ? 1.0f : 0.0f;
  float g_y  = (idx == 7) ? 1.0f : 0.0f;
  float g_1  = (idx == 8) ? 1.0f : 0.0f;

  for (int tile = 0; tile < P; tile += 16) {
    if (lane < 16) {
      int p = tile + lane;
      pt[lane] = make_float4(src[p * 2 + 0], src[p * 2 + 1],
                             dst[p * 2 + 0], dst[p * 2 + 1]);
    }
    if (tile + 16 < P) {
      __builtin_prefetch(src + (tile + 16) * 2, 0, 3);
      __builtin_prefetch(dst + (tile + 16) * 2, 0, 3);
    }
    __syncthreads();
#pragma unroll
    for (int cc = 0; cc < 8; ++cc) {
      float4 pd = pt[cc * 2 + half];          // ds_load_b128
      float t  = g_x * pd.x + g_y * pd.y + g_1;
      float e0 = ce_x * pd.x + ce_y * pd.y + ce_1 - pd.z * t;
      float e1 = co_x * pd.x + co_y * pd.y + co_1 - pd.w * t;
      v2f ab = {e0, e1};
      // AtA accumulation: SRC0 (Aᵀ frag) and SRC1 (A frag) are identical.
      acc = __builtin_amdgcn_wmma_f32_16x16x4_f32(
          false, ab, false, ab, (short)0, acc, false, false);
    }
    __syncthreads();
  }

#pragma unroll
  for (int r = 0; r < 8; ++r) Nsh[r + half * 8][idx] = acc[r];
  __syncthreads();

  if (lane == 0) {
    float a[9][9], Vm[9][9];
    for (int i = 0; i < 9; i++)
      for (int j = 0; j < 9; j++) {
        a[i][j] = Nsh[i][j];
        Vm[i][j] = (i == j) ? 1.0f : 0.0f;
      }
    for (int sweep = 0; sweep < 12; sweep++) {
      for (int p = 0; p < 8; p++) {
        for (int q = p + 1; q < 9; q++) {
          float apq = a[p][q];
          if (fabsf(apq) < 1e-25f) continue;
          float theta = 0.5f * (a[q][q] - a[p][p]) / apq;
          float t = 1.0f / (fabsf(theta) + sqrtf(theta * theta + 1.0f));
          if (theta < 0.0f) t = -t;
          float c = rsqrtf(t * t + 1.0f);
          float s = t * c;
          for (int i = 0; i < 9; i++) {
            float aip = a[i][p], aiq = a[i][q];
            a[i][p] = c * aip - s * aiq;
            a[i][q] = s * aip + c * aiq;
          }
          for (int i = 0; i < 9; i++) {
            float api = a[p][i], aqi = a[q][i];
            a[p][i] = c * api - s * aqi;
            a[q][i] = s * api + c * aqi;
          }
          for (int i = 0; i < 9; i++) {
            float vip = Vm[i][p], viq = Vm[i][q];
            Vm[i][p] = c * vip - s * viq;
            Vm[i][q] = s * vip + c * viq;
          }
        }
      }
    }
    int qm = 0;
    float mn = a[0][0];
    for (int i = 1; i < 9; i++)
      if (a[i][i] < mn) { mn = a[i][i]; qm = i; }
    float* ho = Hout + ((size_t)v * B + b) * 9;
    for (int i = 0; i < 9; i++) ho[i] = Vm[i][qm];
  }
}

// ---------------------------------------------------------------------------
// Kernel 3: masked smooth-L1 reprojection loss, reduced into acc[v], acc[3+v]
// ---------------------------------------------------------------------------
__global__ void loss_kernel(const float* __restrict__ Pgt,
                            const float* __restrict__ Qgt,
                            const float* __restrict__ Hws,
                            float* __restrict__ acc, int B, int M) {
  int v = blockIdx.z;
  int b = blockIdx.y;
  int tid = threadIdx.x;
  int p = blockIdx.x * blockDim.x + tid;
  int P = M * 5;

  __shared__ float Hs[9];
  __shared__ float sl[256];
  __shared__ float sc[256];
  if (tid < 9) Hs[tid] = Hws[((size_t)v * B + b) * 9 + tid];
  __syncthreads();

  float ls = 0.0f, cn = 0.0f;
  if (p < P) {
    size_t fi = (size_t)b * P + p;
    float x = Pgt[fi * 2 + 0], y = Pgt[fi * 2 + 1];
    float n0 = Hs[0] * x + Hs[1] * y + Hs[2];
    float n1 = Hs[3] * x + Hs[4] * y + Hs[5];
    float n2 = Hs[6] * x + Hs[7] * y + Hs[8];
    float inv = 1.0f / (n2 + 1e-10f);
    float rx = n0 * inv, rz = n1 * inv;
    bool msk = (rx > -45.0f) && (rx < 45.0f) && (rz > 0.0f) && (rz < 80.0f);
    float gx = Qgt[fi * 3 + 0], gz = Qgt[fi * 3 + 2];
    float dx = rx - gx, dz = rz - gz;
    float ax = fabsf(dx), az = fabsf(dz);
    float hx = (ax < 1.0f) ? 0.5f * dx * dx : ax - 0.5f;
    float hz = (az < 1.0f) ? 0.5f * dz * dz : az - 0.5f;
    if (msk) { ls = hx + hz; cn = 1.0f; }
  }
  sl[tid] = ls;
  sc[tid] = cn;
  __syncthreads();
  for (int s = 128; s > 0; s >>= 1) {
    if (tid < s) {
      sl[tid] += sl[tid + s];
      sc[tid] += sc[tid + s];
    }
    __syncthreads();
  }
  if (tid == 0) {
    atomicAdd(&acc[v], sl[0]);
    atomicAdd(&acc[3 + v], sc[0]);
  }
}

__global__ void zero_acc_kernel(float* __restrict__ acc) {
  if (threadIdx.x < 8) acc[threadIdx.x] = 0.0f;
}

__global__ void finalize_kernel(const float* __restrict__ acc,
                                float* __restrict__ out) {
  if (threadIdx.x == 0 && blockIdx.x == 0) {
    float s = 0.0f;
    for (int v = 0; v < 3; v++) {
      float cnt = fmaxf(2.0f * acc[3 + v], 1.0f);
      s += acc[v] / cnt;
    }
    out[0] = s * (1.0f / 3.0f);
  }
}

// ---------------------------------------------------------------------------
extern "C" void kernel_launch(void* const* d_in, const int* in_sizes, int n_in,
                              void* d_out, int out_size, void* d_ws,
                              size_t ws_size, hipStream_t stream) {
  (void)n_in; (void)out_size; (void)ws_size;
  const float* center   = (const float*)d_in[1];
  const int*   cls_ids  = (const int*)d_in[2];
  const float* ry       = (const float*)d_in[3];
  const float* size3d   = (const float*)d_in[4];
  const float* Pgt      = (const float*)d_in[5];
  const float* Qgt      = (const float*)d_in[6];
  const float* off_t    = (const float*)d_in[7];
  const float* depth_t  = (const float*)d_in[8];
  const float* off_o    = (const float*)d_in[9];
  const float* depth_o  = (const float*)d_in[10];
  const float* calibs   = (const float*)d_in[11];
  const float* trans_inv= (const float*)d_in[12];

  int B = in_sizes[11] / 12;      // calibs: B*3*4
  int M = in_sizes[0] / B;        // mask_2d: B*M
  int K = B * M;
  int P = M * 5;

  float* ws = (float*)d_ws;
  float* dstws = ws;                              // 3*K*10 floats
  float* Hws   = ws + (size_t)3 * K * 10;         // 3*B*9 floats
  float* acc   = Hws + (size_t)3 * B * 9;         // 8 floats

  build_dst_kernel<<<(K + 255) / 256, 256, 0, stream>>>(
      center, cls_ids, ry, size3d, off_t, depth_t, off_o, depth_o, calibs,
      trans_inv, dstws, K, M);
  zero_acc_kernel<<<1, 32, 0, stream>>>(acc);
  ata_wmma_kernel<<<dim3(B, 3), 32, 0, stream>>>(Pgt, dstws, Hws, B, M);
  loss_kernel<<<dim3((P + 255) / 256, B, 3), 256, 0, stream>>>(Pgt, Qgt, Hws,
                                                               acc, B, M);
  finalize_kernel<<<1, 1, 0, stream>>>(acc, (float*)d_out);
}